// SoftDTW_16776142258552
// MI455X (gfx1250) — compile-verified
//
#include <hip/hip_runtime.h>
#include <hip/hip_bf16.h>

// SoftDTW on MI455X (gfx1250, wave32):
//   Stage 1: row norms x2[i]=||x_i||^2, y2[j]=||y_j||^2
//   Stage 2: D = x2 + y2^T - 2 * x @ y^T via V_WMMA_F32_16X16X4_F32,
//            stored in ANTI-DIAGONAL-MAJOR layout so the serial DP phase
//            reads each diagonal as one contiguous, coalesced run.
//   Stage 3: soft-DTW wavefront DP, single 1024-thread workgroup,
//            3 rolling anti-diagonals in LDS, one barrier per diagonal.
//
// ws layout: Ddiag[2048*2048] f32 (16MB) | x2[2048] | y2[2048]

#define NROWS 2048
#define KDIM  256
#define BIGF  1.0e10f
#define TOT   (NROWS * NROWS)        // 4194304
#define LDIAG (2 * NROWS - 1)        // 4095 anti-diagonals

typedef __attribute__((ext_vector_type(2))) float v2f;
typedef __attribute__((ext_vector_type(8))) float v8f;

// base offset of anti-diagonal dd (0..LDIAG-1) in diagonal-major storage
__device__ __forceinline__ int diag_base(int dd) {
    return (dd < NROWS) ? (dd * (dd + 1)) / 2
                        : TOT - ((LDIAG - dd) * (LDIAG - dd + 1)) / 2;
}
// first row index present on diagonal dd
__device__ __forceinline__ int diag_rlo(int dd) {
    return (dd < NROWS) ? 0 : dd - (NROWS - 1);
}

// ---------------- Stage 1: row squared norms ----------------
__global__ void rownorm_kernel(const float* __restrict__ x,
                               const float* __restrict__ y,
                               float* __restrict__ x2,
                               float* __restrict__ y2) {
    int row = blockIdx.x * blockDim.x + threadIdx.x;
    if (row >= NROWS) return;
    const float4* px = (const float4*)(x + (size_t)row * KDIM);
    const float4* py = (const float4*)(y + (size_t)row * KDIM);
    float sx = 0.f, sy = 0.f;
#pragma unroll 8
    for (int k = 0; k < KDIM / 4; ++k) {
        float4 a = px[k];
        float4 b = py[k];
        sx += a.x * a.x + a.y * a.y + a.z * a.z + a.w * a.w;
        sy += b.x * b.x + b.y * b.y + b.z * b.z + b.w * b.w;
    }
    x2[row] = sx;
    y2[row] = sy;
}

// ---------------- Stage 2: D = x2 + y2^T - 2 x y^T (WMMA f32) ----------------
// One wave32 per 16x16 output tile. A is 16x4 f32 (2 VGPRs/lane), B is 4x16
// f32 (2 VGPRs/lane). Per ISA layout, lane l<16 holds K={k0,k0+1}, lane l>=16
// holds K={k0+2,k0+3}, for row (l&15) of its matrix -> contiguous float2 load.
// Epilogue scatters into anti-diagonal-major D (off the serial critical path).
__global__ __launch_bounds__(256)
void pairwise_dist_wmma(const float* __restrict__ x,
                        const float* __restrict__ y,
                        const float* __restrict__ x2,
                        const float* __restrict__ y2,
                        float* __restrict__ Ddiag) {
    int wave  = (blockIdx.x * blockDim.x + threadIdx.x) >> 5;   // 0..16383
    int lane  = threadIdx.x & 31;
    int tileM = wave >> 7;      // 0..127
    int tileN = wave & 127;     // 0..127
    int half  = lane >> 4;      // 0 or 1 (K-halves)
    int l     = lane & 15;

    const float* pa = x + (size_t)(tileM * 16 + l) * KDIM + half * 2;
    const float* pb = y + (size_t)(tileN * 16 + l) * KDIM + half * 2;

    v8f acc = {0.f, 0.f, 0.f, 0.f, 0.f, 0.f, 0.f, 0.f};
#pragma unroll 8
    for (int k = 0; k < KDIM; k += 4) {
        v2f a = *(const v2f*)(pa + k);
        v2f b = *(const v2f*)(pb + k);
        acc = __builtin_amdgcn_wmma_f32_16x16x4_f32(
            /*neg_a=*/false, a, /*neg_b=*/false, b,
            /*c_mod=*/(short)0, acc, /*reuse_a=*/false, /*reuse_b=*/false);
    }

    // C/D layout: VGPR r, lanes 0-15 -> M=r, N=lane; lanes 16-31 -> M=r+8.
    int n = tileN * 16 + l;
    float yn = y2[n];
#pragma unroll
    for (int r = 0; r < 8; ++r) {
        int m  = tileM * 16 + r + half * 8;
        float val = x2[m] + yn - 2.0f * acc[r];
        int dd = m + n;                                // anti-diagonal index
        Ddiag[diag_base(dd) + (m - diag_rlo(dd))] = val;
    }
}

// ---------------- Stage 3: soft-DTW DP (anti-diagonal wavefront) ----------------
// R is (N+1)x(M+1). Rolling buffers hold diagonal d: buf[i] = R[i, d-i].
// cell (i,j), j=d-i:  diag=prev2[i-1], up=prev[i-1], left=prev[i].
// D for DP diagonal d lives contiguously at Ddiag[diag_base(d-2) + ...].
__global__ __launch_bounds__(1024)
void softdtw_dp(const float* __restrict__ Ddiag, float* __restrict__ out) {
    const int N = NROWS, M = NROWS;
    __shared__ float R[3][2064];
    int tid = threadIdx.x;

    if (tid == 0) {
        R[0][0] = 0.0f;        // R[0,0]
        R[1][0] = BIGF;        // R[0,1]
        R[1][1] = BIGF;        // R[1,0]
    }
    __syncthreads();

    for (int d = 2; d <= N + M; ++d) {
        float* cur   = R[d % 3];
        const float* prev  = R[(d - 1) % 3];
        const float* prev2 = R[(d - 2) % 3];

        if (tid == 0) {
            if (d <= M) cur[0] = BIGF;   // R[0, d]
            if (d <= N) cur[d] = BIGF;   // R[d, 0]
        }

        int ilo = d - M; if (ilo < 1) ilo = 1;
        int ihi = d - 1; if (ihi > N) ihi = N;

        int dd = d - 2;                                   // D's diagonal index
        const float* drow = Ddiag + diag_base(dd) - diag_rlo(dd) - 1;
        // drow[i] == D element for cell (i, d-i): r=i-1 -> base + (r - rlo)

        for (int i = ilo + tid; i <= ihi; i += 1024) {
            float rd = prev2[i - 1];
            float ru = prev[i - 1];
            float rl = prev[i];
            float dv = drow[i];                           // contiguous per diag
            float mn = fminf(rd, fminf(ru, rl));
            // softmin = mn - log(exp(mn-rd)+exp(mn-ru)+exp(mn-rl)), gamma=1
            float s = __expf(mn - rd) + __expf(mn - ru) + __expf(mn - rl);
            cur[i] = dv + (mn - __logf(s));
        }
        __syncthreads();
    }

    if (tid == 0) out[0] = R[(N + M) % 3][N];   // R[N, M]
}

extern "C" void kernel_launch(void* const* d_in, const int* in_sizes, int n_in,
                              void* d_out, int out_size, void* d_ws, size_t ws_size,
                              hipStream_t stream) {
    const float* x = (const float*)d_in[0];
    const float* y = (const float*)d_in[1];
    float* Ddiag = (float*)d_ws;                    // 2048*2048 f32 = 16MB
    float* x2 = Ddiag + (size_t)TOT;                // 2048 f32
    float* y2 = x2 + NROWS;                         // 2048 f32
    float* out = (float*)d_out;

    rownorm_kernel<<<NROWS / 256, 256, 0, stream>>>(x, y, x2, y2);
    // 128x128 tiles of 16x16, one wave each -> 16384 waves, 8 waves/block.
    pairwise_dist_wmma<<<2048, 256, 0, stream>>>(x, y, x2, y2, Ddiag);
    softdtw_dp<<<1, 1024, 0, stream>>>(Ddiag, out);
}